// QuanvolutionHybridModel_65481071408466
// MI455X (gfx1250) — compile-verified
//
#include <hip/hip_runtime.h>
#include <math.h>

typedef float v2f __attribute__((ext_vector_type(2)));
typedef float v8f __attribute__((ext_vector_type(8)));

// ---------------------------------------------------------------------------
// Generic complex 2x2 gate on wire with bit-mask MASK (compile-time constant,
// fully unrolled -> state stays in VGPRs).
// ---------------------------------------------------------------------------
template <int MASK>
__device__ __forceinline__ void gate_c2x2(float* sr, float* si,
                                          float m00r, float m00i,
                                          float m01r, float m01i,
                                          float m10r, float m10i,
                                          float m11r, float m11i) {
#pragma unroll
  for (int i = 0; i < 16; ++i) {
    if (i & MASK) continue;
    const int j = i | MASK;
    const float ar = sr[i], ai = si[i];
    const float br = sr[j], bi = si[j];
    sr[i] = m00r * ar - m00i * ai + m01r * br - m01i * bi;
    si[i] = m00r * ai + m00i * ar + m01r * bi + m01i * br;
    sr[j] = m10r * ar - m10i * ai + m11r * br - m11i * bi;
    si[j] = m10r * ai + m10i * ar + m11r * bi + m11i * br;
  }
}

// Real RY gate for the all-real classifier state.
template <int MASK>
__device__ __forceinline__ void ry_real(float* a, float c, float s) {
#pragma unroll
  for (int i = 0; i < 16; ++i) {
    if (i & MASK) continue;
    const int j = i | MASK;
    const float x = a[i], y = a[j];
    a[i] = c * x - s * y;
    a[j] = s * x + c * y;
  }
}

// ---------------------------------------------------------------------------
// Kernel 1: one thread per (image, patch in {0,1}).  Only patches (0,0) and
// (0,1) of the 14x14 grid feed the classifier -> everything else is dead code.
// 16 complex amplitudes live in registers; writes 4 <Z> values per patch.
// ---------------------------------------------------------------------------
__global__ __launch_bounds__(256) void quanv_patches(
    const float* __restrict__ x, const float* __restrict__ qp,
    float* __restrict__ theta, int B) {
  const int tid = blockIdx.x * blockDim.x + threadIdx.x;
  if (tid >= 2 * B) return;
  const int b = tid >> 1;
  const int p = tid & 1;  // patch column j = p, row i = 0

  const float* img = x + (size_t)b * 784;
  const float t0 = img[0 * 28 + 2 * p + 0];
  const float t1 = img[0 * 28 + 2 * p + 1];
  const float t2 = img[1 * 28 + 2 * p + 0];
  const float t3 = img[1 * 28 + 2 * p + 1];

  // Product state after RY(t_w) on each wire applied to |0000>
  float cs[4][2];
  __sincosf(0.5f * t0, &cs[0][1], &cs[0][0]);
  __sincosf(0.5f * t1, &cs[1][1], &cs[1][0]);
  __sincosf(0.5f * t2, &cs[2][1], &cs[2][0]);
  __sincosf(0.5f * t3, &cs[3][1], &cs[3][0]);

  float sr[16], si[16];
#pragma unroll
  for (int i = 0; i < 16; ++i) {
    sr[i] = cs[0][(i >> 3) & 1] * cs[1][(i >> 2) & 1] *
            cs[2][(i >> 1) & 1] * cs[3][i & 1];
    si[i] = 0.0f;
  }

  float c, s;
  // RX(qp0) on wire 0 (mask 8): [[c,-is],[-is,c]]
  __sincosf(0.5f * qp[0], &s, &c);
  gate_c2x2<8>(sr, si, c, 0.f, 0.f, -s, 0.f, -s, c, 0.f);
  // RY(qp1) on wire 1 (mask 4)
  __sincosf(0.5f * qp[1], &s, &c);
  gate_c2x2<4>(sr, si, c, 0.f, -s, 0.f, s, 0.f, c, 0.f);
  // RZ(qp2) on wire 2 (mask 2): diag(e^{-it/2}, e^{+it/2})
  __sincosf(0.5f * qp[2], &s, &c);
  gate_c2x2<2>(sr, si, c, -s, 0.f, 0.f, 0.f, 0.f, c, s);
  // CNOT(0,1): for indices with bit3 set, swap bit2  -> swap (i, i^4), i>=8
#pragma unroll
  for (int i = 8; i < 12; ++i) {
    float tr = sr[i]; sr[i] = sr[i + 4]; sr[i + 4] = tr;
    float ti = si[i]; si[i] = si[i + 4]; si[i + 4] = ti;
  }
  // RY(qp3) on wire 3 (mask 1)
  __sincosf(0.5f * qp[3], &s, &c);
  gate_c2x2<1>(sr, si, c, 0.f, -s, 0.f, s, 0.f, c, 0.f);
  // CNOT(2,3): for indices with bit1 set, swap bit0 -> swap (i, i^1)
#pragma unroll
  for (int i = 2; i < 16; i += 4) {
    float tr = sr[i]; sr[i] = sr[i + 1]; sr[i + 1] = tr;
    float ti = si[i]; si[i] = si[i + 1]; si[i + 1] = ti;
  }
  // RX(qp4) on wire 1 (mask 4)
  __sincosf(0.5f * qp[4], &s, &c);
  gate_c2x2<4>(sr, si, c, 0.f, 0.f, -s, 0.f, -s, c, 0.f);
  // RZ(qp5) on wire 3 (mask 1)
  __sincosf(0.5f * qp[5], &s, &c);
  gate_c2x2<1>(sr, si, c, -s, 0.f, 0.f, 0.f, 0.f, c, s);

  // <Z_w> = sum_i |amp_i|^2 * (1 - 2*bit_w(i))
  float e0 = 0.f, e1 = 0.f, e2 = 0.f, e3 = 0.f;
#pragma unroll
  for (int i = 0; i < 16; ++i) {
    const float pr = sr[i] * sr[i] + si[i] * si[i];
    e0 += (i & 8) ? -pr : pr;
    e1 += (i & 4) ? -pr : pr;
    e2 += (i & 2) ? -pr : pr;
    e3 += (i & 1) ? -pr : pr;
  }
  float* o = theta + (size_t)tid * 4;  // = theta[b*8 + p*4 + w]
  o[0] = e0; o[1] = e1; o[2] = e2; o[3] = e3;
}

// ---------------------------------------------------------------------------
// Kernel 2: classifier circuit (all-real amplitudes) + WMMA readout GEMM
// (expv[16x4] @ W^T[4x16]) + log-softmax.  One wave handles 16 images with a
// single v_wmma_f32_16x16x4_f32.
// ---------------------------------------------------------------------------
__global__ __launch_bounds__(256) void classifier_readout(
    const float* __restrict__ theta, const float* __restrict__ rw,
    const float* __restrict__ rb, float* __restrict__ out, int B) {
  __shared__ float tile[8 * 16 * 17];  // 8 waves/block, 16x16 tile, pitch 17

  const int lane = threadIdx.x & 31;
  const int wv = threadIdx.x >> 5;
  const int waveId = blockIdx.x * 8 + wv;
  const int base = waveId * 16;
  const int m = lane & 15;    // row within tile (image)
  const int hi = lane >> 4;   // half-wave: selects K pair {0,1} vs {2,3}
  const int img = base + m;
  const int imgc = img < B ? img : (B - 1);  // clamp loads; no EXEC divergence

  // ---- classifier circuit: real 16-amplitude state, lanes 16-31 mirror 0-15
  const float* th = theta + (size_t)imgc * 8;
  float a[16];
#pragma unroll
  for (int i = 0; i < 16; ++i) a[i] = 0.0f;
  a[0] = 1.0f;

#pragma unroll
  for (int layer = 0; layer < 2; ++layer) {
    float c, s;
    __sincosf(0.5f * th[layer * 4 + 0], &s, &c); ry_real<8>(a, c, s);
    __sincosf(0.5f * th[layer * 4 + 1], &s, &c); ry_real<4>(a, c, s);
    __sincosf(0.5f * th[layer * 4 + 2], &s, &c); ry_real<2>(a, c, s);
    __sincosf(0.5f * th[layer * 4 + 3], &s, &c); ry_real<1>(a, c, s);
    // CZ(0,1), CZ(1,2), CZ(2,3): negate amplitudes where both bits set
#pragma unroll
    for (int i = 0; i < 16; ++i) {
      if (((i & 8) && (i & 4)) ^ ((i & 4) && (i & 2)) ^ ((i & 2) && (i & 1)))
        a[i] = -a[i];
    }
  }

  float e0 = 0.f, e1 = 0.f, e2 = 0.f, e3 = 0.f;
#pragma unroll
  for (int i = 0; i < 16; ++i) {
    const float pr = a[i] * a[i];
    e0 += (i & 8) ? -pr : pr;
    e1 += (i & 4) ? -pr : pr;
    e2 += (i & 2) ? -pr : pr;
    e3 += (i & 1) ? -pr : pr;
  }

  // ---- A fragment (16x4 f32): lanes 0-15 -> K=0,1 ; lanes 16-31 -> K=2,3
  v2f A;
  A.x = hi ? e2 : e0;
  A.y = hi ? e3 : e1;

  // ---- B fragment (4x16 f32): B[k][n] = readout_w[n][k], zero-pad n>=10
  const int nc = m < 10 ? m : 9;
  const float b0 = rw[nc * 4 + 2 * hi + 0];
  const float b1 = rw[nc * 4 + 2 * hi + 1];
  v2f Bf;
  Bf.x = (m < 10) ? b0 : 0.0f;
  Bf.y = (m < 10) ? b1 : 0.0f;

  v8f C = {0.f, 0.f, 0.f, 0.f, 0.f, 0.f, 0.f, 0.f};
  // D = A x B + 0 : logits tile, 16 images x 16 (10 used) classes
  C = __builtin_amdgcn_wmma_f32_16x16x4_f32(
      /*neg_a=*/false, A, /*neg_b=*/false, Bf,
      /*c_mod=*/(short)0, C, /*reuse_a=*/false, /*reuse_b=*/false);

  // ---- spill D tile to LDS (rows: lanes 0-15 -> M=0..7, 16-31 -> M=8..15)
  float* tw = tile + wv * (16 * 17);
  const int rowBase = hi * 8;
#pragma unroll
  for (int r = 0; r < 8; ++r) tw[(rowBase + r) * 17 + m] = C[r];
  __syncthreads();

  // ---- per-row bias + log-softmax, 10 outputs per image
  if (lane < 16 && img < B) {
    float lg[10];
    float mx = -3.4e38f;
#pragma unroll
    for (int n = 0; n < 10; ++n) {
      lg[n] = tw[m * 17 + n] + rb[n];
      mx = fmaxf(mx, lg[n]);
    }
    float sum = 0.f;
#pragma unroll
    for (int n = 0; n < 10; ++n) sum += __expf(lg[n] - mx);
    const float lse = mx + __logf(sum);
    float* o = out + (size_t)img * 10;
#pragma unroll
    for (int n = 0; n < 10; ++n) o[n] = lg[n] - lse;
  }
}

// ---------------------------------------------------------------------------
extern "C" void kernel_launch(void* const* d_in, const int* in_sizes, int n_in,
                              void* d_out, int out_size, void* d_ws,
                              size_t ws_size, hipStream_t stream) {
  const float* x = (const float*)d_in[0];   // (B, 28, 28)
  const float* qp = (const float*)d_in[1];  // (6,)
  const float* rw = (const float*)d_in[2];  // (10, 4)
  const float* rb = (const float*)d_in[3];  // (10,)
  float* out = (float*)d_out;               // (B, 10)
  float* theta = (float*)d_ws;              // (B, 8) scratch

  const int B = in_sizes[0] / 784;

  const int nPatch = 2 * B;
  quanv_patches<<<(nPatch + 255) / 256, 256, 0, stream>>>(x, qp, theta, B);

  const int nWaves = (B + 15) / 16;
  const int blocks2 = (nWaves + 7) / 8;
  classifier_readout<<<blocks2, 256, 0, stream>>>(theta, rw, rb, out, B);
}